// LocalU1LCNN_72249939853691
// MI455X (gfx1250) — compile-verified
//
#include <hip/hip_runtime.h>
#include <math.h>

// U(1) lattice CNN on MI455X (gfx1250).
// Channel matmuls use V_WMMA_F32_16X16X4_F32 (fp32 WMMA, no precision loss).
// B = 64, L = 128, H = 16, O = 8.

typedef __attribute__((ext_vector_type(2))) float v2f;
typedef __attribute__((ext_vector_type(8))) float v8f;

#define LSZ   128
#define LMASK 127
#define NBATCH 64
#define EPSF  1e-12f

__device__ __forceinline__ int xoff(int b, int c, int i, int j) {
    return ((b * 4 + c) * LSZ + i) * LSZ + j;
}

// ---------------------------------------------------------------------------
// Kernel A: Wilson seed.  seed(b,i,j) = conj( prod_{k<i} ux[b,k,0] * prod_{k<j} uy[b,i,k] )
// One block per batch; thread 0 does the x-axis scan into LDS, then each
// thread owns one row i and scans along j (contiguous loads).
// ---------------------------------------------------------------------------
__global__ void seed_kernel(const float* __restrict__ x,
                            float* __restrict__ seed_r, float* __restrict__ seed_i) {
    __shared__ float axr[LSZ], axi[LSZ];
    int b = blockIdx.x;
    if (threadIdx.x == 0) {
        float cr = 1.f, ci = 0.f;
        axr[0] = cr; axi[0] = ci;
        for (int k = 0; k < LSZ - 1; ++k) {
            float ur = x[xoff(b, 0, k, 0)];
            float ui = x[xoff(b, 1, k, 0)];
            float nr = cr * ur - ci * ui;
            float ni = cr * ui + ci * ur;
            cr = nr; ci = ni;
            axr[k + 1] = cr; axi[k + 1] = ci;
        }
    }
    __syncthreads();
    int i = threadIdx.x;
    float cr = axr[i], ci = axi[i];
    int base = (b * LSZ + i) * LSZ;
    seed_r[base] = cr; seed_i[base] = -ci;   // conjugate on store
    for (int k = 0; k < LSZ - 1; ++k) {
        float ur = x[xoff(b, 2, i, k)];
        float ui = x[xoff(b, 3, i, k)];
        float nr = cr * ur - ci * ui;
        float ni = cr * ui + ci * ur;
        cr = nr; ci = ni;
        seed_r[base + k + 1] = cr;
        seed_i[base + k + 1] = -ci;
    }
}

// ---------------------------------------------------------------------------
// Kernel B: pc = cos(plaquette).  cos() is 2*pi periodic so the wrap is a no-op.
// ---------------------------------------------------------------------------
__global__ void plaq_kernel(const float* __restrict__ x, float* __restrict__ pc) {
    int idx = blockIdx.x * blockDim.x + threadIdx.x;   // over B*L*L
    int j = idx & LMASK;
    int i = (idx >> 7) & LMASK;
    int b = idx >> 14;
    int i1 = (i + 1) & LMASK, j1 = (j + 1) & LMASK;
    float ax    = atan2f(x[xoff(b, 1, i, j)],  x[xoff(b, 0, i, j)]);
    float ax_j1 = atan2f(x[xoff(b, 1, i, j1)], x[xoff(b, 0, i, j1)]);
    float ay    = atan2f(x[xoff(b, 3, i, j)],  x[xoff(b, 2, i, j)]);
    float ay_i1 = atan2f(x[xoff(b, 3, i1, j)], x[xoff(b, 2, i1, j)]);
    pc[idx] = cosf(ax + ay_i1 - ax_j1 - ay);
}

// ---------------------------------------------------------------------------
// Kernel C: initial features (built on the fly from seed/pc) -> lconv1 -> modReLU.
// One wave32 per 16-site strip along j. WMMA mapping:
//   A (16x4 fp32)  = per-tap weight block, M = 16 out channels
//   B (4x16 fp32)  = transported features, N = 16 sites, K = {3 ch + zero pad}
//   C/D (16x16)    = accumulators, 2 WMMA per tap (real/imag), 10 per tile.
// Lanes l and l+16 compute features for the same site (l&15), so the B
// fragment (K rows 0,1 lo-half / 2,3 hi-half) needs no cross-lane traffic.
// Output g stored interleaved [b][i][j][ch] for contiguous layer-2 loads.
// ---------------------------------------------------------------------------
__global__ __launch_bounds__(256) void lconv1_kernel(
    const float* __restrict__ x,
    const float* __restrict__ seed_r, const float* __restrict__ seed_i,
    const float* __restrict__ pc,
    const float* __restrict__ center1, const float* __restrict__ fwd_w1,
    const float* __restrict__ bwd_w1,  const float* __restrict__ bias1,
    float* __restrict__ g_r, float* __restrict__ g_i) {
    __shared__ float Wl[16 * 20];   // [out ch][tap*4 + c], c=3 zero pad
    for (int t = threadIdx.x; t < 320; t += 256) {
        int o = t / 20, col = t % 20;
        int tap = col >> 2, c = col & 3;
        float v = 0.f;
        if (c < 3) {
            if      (tap == 0) v = center1[o * 3 + c];
            else if (tap == 1) v = fwd_w1[o * 3 + c];            // mu = 0
            else if (tap == 2) v = fwd_w1[48 + o * 3 + c];       // mu = 1
            else if (tap == 3) v = bwd_w1[o * 3 + c];
            else               v = bwd_w1[48 + o * 3 + c];
        }
        Wl[t] = v;
    }
    __syncthreads();

    int lane = threadIdx.x & 31;
    int wave = threadIdx.x >> 5;
    int tile = blockIdx.x * 8 + wave;          // 65536 tiles total
    int j0 = (tile & 7) << 4;
    int i  = (tile >> 3) & LMASK;
    int b  = tile >> 10;
    int n  = lane & 15;
    int hi = (lane >= 16) ? 1 : 0;
    int j  = j0 + n;
    int M  = n;
    int pbase = b * (LSZ * LSZ);

    v8f accr = {0.f, 0.f, 0.f, 0.f, 0.f, 0.f, 0.f, 0.f};
    v8f acci = {0.f, 0.f, 0.f, 0.f, 0.f, 0.f, 0.f, 0.f};

#pragma unroll
    for (int tap = 0; tap < 5; ++tap) {
        int si = i, sj = j;
        float lr = 1.f, li = 0.f;
        if (tap == 1) { si = (i + 1) & LMASK; lr = x[xoff(b, 0, i, j)];  li =  x[xoff(b, 1, i, j)]; }
        else if (tap == 2) { sj = (j + 1) & LMASK; lr = x[xoff(b, 2, i, j)];  li =  x[xoff(b, 3, i, j)]; }
        else if (tap == 3) { si = (i - 1) & LMASK; lr = x[xoff(b, 0, si, j)]; li = -x[xoff(b, 1, si, j)]; }
        else if (tap == 4) { sj = (j - 1) & LMASK; lr = x[xoff(b, 2, i, sj)]; li = -x[xoff(b, 3, i, sj)]; }
        float sr = seed_r[(b * LSZ + si) * LSZ + sj];
        float sv = seed_i[(b * LSZ + si) * LSZ + sj];
        float tr = lr * sr - li * sv;           // transported complex seed
        float ti = lr * sv + li * sr;
        float p0  = pc[pbase + si * LSZ + sj];
        float pim = pc[pbase + ((si - 1) & LMASK) * LSZ + sj];
        float pip = pc[pbase + ((si + 1) & LMASK) * LSZ + sj];
        float pjm = pc[pbase + si * LSZ + ((sj - 1) & LMASK)];
        float pjp = pc[pbase + si * LSZ + ((sj + 1) & LMASK)];
        float pav = (p0 + pim + pip + pjm + pjp) * 0.2f;
        // channel scalars {1, pc, pavg}: features = t * scalar_c
        float fr0 = tr, fr1 = tr * p0, fr2 = tr * pav;
        float fi0 = ti, fi1 = ti * p0, fi2 = ti * pav;

        const float* wrow = &Wl[M * 20 + tap * 4];
        float a0 = hi ? wrow[2] : wrow[0];
        float a1 = hi ? wrow[3] : wrow[1];
        float br0 = hi ? fr2 : fr0,  br1 = hi ? 0.f : fr1;
        float bi0 = hi ? fi2 : fi0,  bi1 = hi ? 0.f : fi1;
        v2f a  = {a0, a1};
        v2f bR = {br0, br1};
        v2f bI = {bi0, bi1};
        accr = __builtin_amdgcn_wmma_f32_16x16x4_f32(false, a, false, bR, (short)0, accr, false, false);
        acci = __builtin_amdgcn_wmma_f32_16x16x4_f32(false, a, false, bI, (short)0, acci, false, false);
    }

    // modReLU + store interleaved [b][i][j][ch]
    int gbase = ((b * LSZ + i) * LSZ + j) * 16 + hi * 8;
#pragma unroll
    for (int v = 0; v < 8; ++v) {
        int ch = v + hi * 8;
        float r = accr[v], im = acci[v];
        float mag = sqrtf(r * r + im * im + EPSF);
        float act = fmaxf(mag + bias1[ch], 0.f);
        float sc  = act / fmaxf(mag, EPSF);
        g_r[gbase + v] = r * sc;
        g_i[gbase + v] = im * sc;
    }
}

// ---------------------------------------------------------------------------
// Kernel D: lconv2 (16->8, padded to 16 rows) -> modReLU -> readout, fused.
// Per tile: 5 taps x 4 K-steps x 2 components = 40 V_WMMA_F32_16X16X4_F32.
// g loads are 16 contiguous floats per (site,component) -> b128 loads, L2 hits.
// ---------------------------------------------------------------------------
__global__ __launch_bounds__(256) void lconv2_kernel(
    const float* __restrict__ x,
    const float* __restrict__ g_r, const float* __restrict__ g_i,
    const float* __restrict__ center2, const float* __restrict__ fwd_w2,
    const float* __restrict__ bwd_w2,  const float* __restrict__ bias2,
    const float* __restrict__ ro_w,    const float* __restrict__ ro_b,
    float* __restrict__ out) {
    __shared__ float Wl[16 * 80];   // [out ch (8 used, 8 zero)][tap*16 + c]
    for (int t = threadIdx.x; t < 1280; t += 256) {
        int o = t / 80, col = t % 80;
        int tap = col >> 4, c = col & 15;
        float v = 0.f;
        if (o < 8) {
            if      (tap == 0) v = center2[o * 16 + c];
            else if (tap == 1) v = fwd_w2[o * 16 + c];
            else if (tap == 2) v = fwd_w2[128 + o * 16 + c];
            else if (tap == 3) v = bwd_w2[o * 16 + c];
            else               v = bwd_w2[128 + o * 16 + c];
        }
        Wl[t] = v;
    }
    __syncthreads();

    int lane = threadIdx.x & 31;
    int wave = threadIdx.x >> 5;
    int tile = blockIdx.x * 8 + wave;
    int j0 = (tile & 7) << 4;
    int i  = (tile >> 3) & LMASK;
    int b  = tile >> 10;
    int n  = lane & 15;
    int hi = (lane >= 16) ? 1 : 0;
    int j  = j0 + n;
    int M  = n;

    v8f accr = {0.f, 0.f, 0.f, 0.f, 0.f, 0.f, 0.f, 0.f};
    v8f acci = {0.f, 0.f, 0.f, 0.f, 0.f, 0.f, 0.f, 0.f};
    float fr[16], fi[16];

#pragma unroll
    for (int tap = 0; tap < 5; ++tap) {
        int si = i, sj = j;
        float lr = 1.f, li = 0.f;
        if (tap == 1) { si = (i + 1) & LMASK; lr = x[xoff(b, 0, i, j)];  li =  x[xoff(b, 1, i, j)]; }
        else if (tap == 2) { sj = (j + 1) & LMASK; lr = x[xoff(b, 2, i, j)];  li =  x[xoff(b, 3, i, j)]; }
        else if (tap == 3) { si = (i - 1) & LMASK; lr = x[xoff(b, 0, si, j)]; li = -x[xoff(b, 1, si, j)]; }
        else if (tap == 4) { sj = (j - 1) & LMASK; lr = x[xoff(b, 2, i, sj)]; li = -x[xoff(b, 3, i, sj)]; }
        const float* grp = &g_r[((b * LSZ + si) * LSZ + sj) * 16];
        const float* gip = &g_i[((b * LSZ + si) * LSZ + sj) * 16];
#pragma unroll
        for (int c = 0; c < 16; ++c) {
            float gr = grp[c], gg = gip[c];
            fr[c] = lr * gr - li * gg;       // link * g (complex)
            fi[c] = lr * gg + li * gr;
        }
        const float* wrow = &Wl[M * 80 + tap * 16];
#pragma unroll
        for (int st = 0; st < 4; ++st) {
            int k0 = st * 4;
            float a0 = hi ? wrow[k0 + 2] : wrow[k0 + 0];
            float a1 = hi ? wrow[k0 + 3] : wrow[k0 + 1];
            float r0 = hi ? fr[k0 + 2] : fr[k0 + 0], r1 = hi ? fr[k0 + 3] : fr[k0 + 1];
            float q0 = hi ? fi[k0 + 2] : fi[k0 + 0], q1 = hi ? fi[k0 + 3] : fi[k0 + 1];
            v2f a  = {a0, a1};
            v2f bR = {r0, r1};
            v2f bI = {q0, q1};
            accr = __builtin_amdgcn_wmma_f32_16x16x4_f32(false, a, false, bR, (short)0, accr, false, false);
            acci = __builtin_amdgcn_wmma_f32_16x16x4_f32(false, a, false, bI, (short)0, acci, false, false);
        }
    }

    // rows 8..15 of A are zero-padded, so lanes 0..15 hold all 8 real channels.
    if (!hi) {
        float sum = 0.f;
#pragma unroll
        for (int v = 0; v < 8; ++v) {
            float r = accr[v], im = acci[v];
            float mag = sqrtf(r * r + im * im + EPSF);
            float act = fmaxf(mag + bias2[v], 0.f);
            float sc  = act / fmaxf(mag, EPSF);
            r *= sc; im *= sc;
            sum += ro_w[v] * (r * r + im * im);
        }
        out[(b * LSZ + i) * LSZ + j] = sum + ro_b[0];
    }
}

// ---------------------------------------------------------------------------
// Launch. Workspace layout (floats): seed_r | seed_i | pc | g_r(16ch) | g_i(16ch)
//   = (1 + 1 + 1 + 16 + 16) * 1,048,576 * 4 B  ~= 140 MB.
// ---------------------------------------------------------------------------
extern "C" void kernel_launch(void* const* d_in, const int* in_sizes, int n_in,
                              void* d_out, int out_size, void* d_ws, size_t ws_size,
                              hipStream_t stream) {
    const float* x       = (const float*)d_in[0];
    const float* center1 = (const float*)d_in[1];
    const float* fwd_w1  = (const float*)d_in[2];
    const float* bwd_w1  = (const float*)d_in[3];
    const float* bias1   = (const float*)d_in[4];
    const float* center2 = (const float*)d_in[5];
    const float* fwd_w2  = (const float*)d_in[6];
    const float* bwd_w2  = (const float*)d_in[7];
    const float* bias2   = (const float*)d_in[8];
    const float* ro_w    = (const float*)d_in[9];
    const float* ro_b    = (const float*)d_in[10];
    float* out = (float*)d_out;

    float* ws = (float*)d_ws;
    const size_t NS = (size_t)NBATCH * LSZ * LSZ;   // 1,048,576 sites
    float* seed_r = ws;
    float* seed_i = seed_r + NS;
    float* pcbuf  = seed_i + NS;
    float* g_r    = pcbuf + NS;
    float* g_i    = g_r + NS * 16;

    seed_kernel<<<NBATCH, LSZ, 0, stream>>>(x, seed_r, seed_i);
    plaq_kernel<<<(int)(NS / 256), 256, 0, stream>>>(x, pcbuf);
    lconv1_kernel<<<8192, 256, 0, stream>>>(x, seed_r, seed_i, pcbuf,
                                            center1, fwd_w1, bwd_w1, bias1, g_r, g_i);
    lconv2_kernel<<<8192, 256, 0, stream>>>(x, g_r, g_i,
                                            center2, fwd_w2, bwd_w2, bias2, ro_w, ro_b, out);
}